// GATNodeClassifier_80711025427228
// MI455X (gfx1250) — compile-verified
//
#include <hip/hip_runtime.h>
#include <math.h>

#define NNODES 100000
#define NEDGES 1600000
#define ETOT   (NEDGES + NNODES)
#define IN_CH  128
#define HID    32
#define NHEADS 4
#define OUT_CH 40
#define NEG_SLOPE 0.2f
#define BN_EPS 1e-5f
#define SM_EPS 1e-16f

typedef float v2f __attribute__((ext_vector_type(2)));
typedef float v8f __attribute__((ext_vector_type(8)));

// ---------------------------------------------------------------------------
// fp32 WMMA GEMM: Y[nrows x ncols] = X[nrows x K] @ W[K x ncols]
// One wave computes one 16x16 output tile via V_WMMA_F32_16X16X4_F32 (K step 4).
// A layout: lane = m + 16*(k/2), vgpr = k%2.  B layout: lane = n + 16*(k/2),
// vgpr = k%2.  C/D: vgpr r -> row r + 8*(lane/16), col = lane%16.
// Edge tiles: clamp addresses (branchless) so EXEC stays all-ones at the WMMA.
// ---------------------------------------------------------------------------
__global__ void gemm16_wmma_f32(const float* __restrict__ X, const float* __restrict__ W,
                                float* __restrict__ Y, int nrows, int K, int ncols) {
  const int lane  = threadIdx.x & 31;
  const int wslot = threadIdx.x >> 5;
  const int tiles_n = (ncols + 15) >> 4;
  const int tiles_m = (nrows + 15) >> 4;
  const int wid = blockIdx.x * (blockDim.x >> 5) + wslot;
  if (wid >= tiles_m * tiles_n) return;   // wave-uniform exit
  const int tm = wid / tiles_n;
  const int tn = wid % tiles_n;
  const int r   = lane & 15;
  const int khi = lane >> 4;              // 0: k%4 in {0,1}; 1: k%4 in {2,3}

  int gm = tm * 16 + r; gm = (gm < nrows) ? gm : (nrows - 1);
  int gn = tn * 16 + r; gn = (gn < ncols) ? gn : (ncols - 1);

  const float* xrow = X + (size_t)gm * K;
  v8f acc = {};
#pragma unroll 4
  for (int k0 = 0; k0 < K; k0 += 4) {
    const int k = k0 + 2 * khi;
    v2f a; a.x = xrow[k];
           a.y = xrow[k + 1];
    v2f b; b.x = W[(size_t)k * ncols + gn];
           b.y = W[(size_t)(k + 1) * ncols + gn];
    acc = __builtin_amdgcn_wmma_f32_16x16x4_f32(false, a, false, b,
                                                (short)0, acc, false, false);
  }

  const int col = tn * 16 + r;
  if (col < ncols) {
#pragma unroll
    for (int i = 0; i < 8; ++i) {
      const int row = tm * 16 + khi * 8 + i;
      if (row < nrows) Y[(size_t)row * ncols + col] = acc[i];
    }
  }
}

// per-node, per-head attention scalars: a_s = <xh[n,h,:], asrc[h,:]>, same for a_d
__global__ void scores_kernel(const float* __restrict__ xh,
                              const float* __restrict__ asrc, const float* __restrict__ adst,
                              float* __restrict__ a_s, float* __restrict__ a_d,
                              int n, int H, int OC) {
  int t = blockIdx.x * blockDim.x + threadIdx.x;
  if (t >= n * H) return;
  const int node = t / H, h = t % H;
  const float* v = xh + (size_t)node * H * OC + (size_t)h * OC;
  float ss = 0.f, sd = 0.f;
  for (int c = 0; c < OC; ++c) {
    ss += v[c] * asrc[h * OC + c];
    sd += v[c] * adst[h * OC + c];
  }
  a_s[t] = ss;
  a_d[t] = sd;
}

// zero the accumulation buffer; m = -inf, den = 0
__global__ void init_kernel(float* __restrict__ accum, int accum_n,
                            float* __restrict__ m, float* __restrict__ den, int nh) {
  int t = blockIdx.x * blockDim.x + threadIdx.x;
  if (t < accum_n) accum[t] = 0.f;
  if (t < nh) { m[t] = __int_as_float(0xff800000); den[t] = 0.f; }
}

__device__ __forceinline__ void atomicMaxFloat(float* addr, float val) {
  if (val >= 0.0f) atomicMax((int*)addr, __float_as_int(val));
  else             atomicMin((unsigned int*)addr, __float_as_uint(val));
}

// e = leaky_relu(a_s[src] + a_d[dst]); store e; segment-max into m[dst]
__global__ void edge_e_kernel(const int* __restrict__ esrc, const int* __restrict__ edst,
                              int E, int Et,
                              const float* __restrict__ a_s, const float* __restrict__ a_d,
                              float* __restrict__ ebuf, float* __restrict__ m, int H) {
  int t = blockIdx.x * blockDim.x + threadIdx.x;
  if (t >= Et * H) return;
  const int edge = t / H, h = t % H;
  int s, d;
  if (edge < E) { s = esrc[edge]; d = edst[edge]; } else { s = d = edge - E; }
  float e = a_s[s * H + h] + a_d[d * H + h];
  e = (e > 0.f) ? e : NEG_SLOPE * e;
  ebuf[t] = e;
  atomicMaxFloat(&m[d * H + h], e);
}

// ex = exp(e - m[dst]); store ex (in place); segment-sum into den[dst]
__global__ void edge_exp_kernel(const int* __restrict__ edst, int E, int Et,
                                float* __restrict__ ebuf,
                                const float* __restrict__ m, float* __restrict__ den, int H) {
  int t = blockIdx.x * blockDim.x + threadIdx.x;
  if (t >= Et * H) return;
  const int edge = t / H, h = t % H;
  const int d = (edge < E) ? edst[edge] : (edge - E);
  const float ex = expf(ebuf[t] - m[d * H + h]);
  ebuf[t] = ex;
  atomicAdd(&den[d * H + h], ex);
}

// alpha = ex / (den[dst] + eps)  (in place; buffer is the d_out alpha section)
__global__ void edge_alpha_kernel(const int* __restrict__ edst, int E, int Et,
                                  float* __restrict__ ebuf,
                                  const float* __restrict__ den, int H) {
  int t = blockIdx.x * blockDim.x + threadIdx.x;
  if (t >= Et * H) return;
  const int edge = t / H, h = t % H;
  const int d = (edge < E) ? edst[edge] : (edge - E);
  ebuf[t] = ebuf[t] / (den[d * H + h] + SM_EPS);
}

// accum[dst, :] += xh[src, :] * alpha[edge, head]; 4 channels per thread
__global__ void scatter_kernel(const int* __restrict__ esrc, const int* __restrict__ edst,
                               int E, int Et,
                               const float* __restrict__ xh, const float* __restrict__ alpha,
                               float* __restrict__ accum, int H, int OC) {
  const int CH = H * OC;
  const int quads = CH >> 2;
  long long t = (long long)blockIdx.x * blockDim.x + threadIdx.x;
  if (t >= (long long)Et * quads) return;
  const int edge = (int)(t / quads);
  const int c = ((int)(t % quads)) << 2;
  const int h = c / OC;
  int s, d;
  if (edge < E) { s = esrc[edge]; d = edst[edge]; } else { s = d = edge - E; }
  const float4 xv = *(const float4*)(xh + (size_t)s * CH + c);
  const float  al = alpha[(size_t)edge * H + h];
  float* dp = accum + (size_t)d * CH + c;
  atomicAdd(dp + 0, xv.x * al);
  atomicAdd(dp + 1, xv.y * al);
  atomicAdd(dp + 2, xv.z * al);
  atomicAdd(dp + 3, xv.w * al);
}

// h = BN(accum + bias); optional ELU
__global__ void bn_kernel(const float* __restrict__ accum, const float* __restrict__ bias,
                          const float* __restrict__ g, const float* __restrict__ be,
                          const float* __restrict__ rm, const float* __restrict__ rv,
                          float* __restrict__ out, int n, int D, int do_elu) {
  int t = blockIdx.x * blockDim.x + threadIdx.x;
  if (t >= n * D) return;
  const int c = t % D;
  float v = accum[t] + bias[c];
  v = (v - rm[c]) * rsqrtf(rv[c] + BN_EPS) * g[c] + be[c];
  if (do_elu) v = (v > 0.f) ? v : expm1f(v);
  out[t] = v;
}

extern "C" void kernel_launch(void* const* d_in, const int* in_sizes, int n_in,
                              void* d_out, int out_size, void* d_ws, size_t ws_size,
                              hipStream_t stream) {
  const float* x    = (const float*)d_in[0];
  const int*   ei   = (const int*)d_in[1];
  const int*   esrc = ei;
  const int*   edst = ei + NEDGES;

  const float* W [3] = {(const float*)d_in[2],  (const float*)d_in[10], (const float*)d_in[18]};
  const float* As[3] = {(const float*)d_in[3],  (const float*)d_in[11], (const float*)d_in[19]};
  const float* Ad[3] = {(const float*)d_in[4],  (const float*)d_in[12], (const float*)d_in[20]};
  const float* Bi[3] = {(const float*)d_in[5],  (const float*)d_in[13], (const float*)d_in[21]};
  const float* Ga[3] = {(const float*)d_in[6],  (const float*)d_in[14], (const float*)d_in[22]};
  const float* Be[3] = {(const float*)d_in[7],  (const float*)d_in[15], (const float*)d_in[23]};
  const float* Rm[3] = {(const float*)d_in[8],  (const float*)d_in[16], (const float*)d_in[24]};
  const float* Rv[3] = {(const float*)d_in[9],  (const float*)d_in[17], (const float*)d_in[25]};
  const int Hh[3] = {NHEADS, NHEADS, 1};
  const int Oc[3] = {HID, HID, OUT_CH};

  float* out  = (float*)d_out;
  float* ws   = (float*)d_ws;
  float* bufA = ws;                                  // xh (N x 128 max)
  float* bufB = bufA + (size_t)NNODES * 128;         // accum / h ping buffer
  float* mbuf = bufB + (size_t)NNODES * 128;         // segment max   (N x H)
  float* den  = mbuf + (size_t)NNODES * NHEADS;      // softmax denom (N x H)
  float* a_s  = den  + (size_t)NNODES * NHEADS;      // src scores    (N x H)
  float* a_d  = a_s  + (size_t)NNODES * NHEADS;      // dst scores    (N x H)

  // alpha sections of d_out double as scratch for e -> exp -> alpha
  float* alphaSec[3];
  alphaSec[0] = out + (size_t)NNODES * OUT_CH;
  alphaSec[1] = alphaSec[0] + (size_t)ETOT * NHEADS;
  alphaSec[2] = alphaSec[1] + (size_t)ETOT * NHEADS;

  const float* hin = x;
  for (int L = 0; L < 3; ++L) {
    const int H = Hh[L], OC = Oc[L], CH = H * OC;

    // 1) xh = hin @ W   (K is 128 for every layer)
    {
      const int tiles = ((NNODES + 15) / 16) * ((CH + 15) / 16);
      gemm16_wmma_f32<<<dim3((tiles + 3) / 4), dim3(128), 0, stream>>>(
          hin, W[L], bufA, NNODES, IN_CH, CH);
    }
    // 2) per-node head scores
    {
      const int nh = NNODES * H;
      scores_kernel<<<dim3((nh + 255) / 256), dim3(256), 0, stream>>>(
          bufA, As[L], Ad[L], a_s, a_d, NNODES, H, OC);
    }
    // 3) init accumulators (GEMM already consumed hin, so bufB is reusable)
    {
      const int an = NNODES * CH;
      const int nh = NNODES * H;
      const int mx = (an > nh) ? an : nh;
      init_kernel<<<dim3((mx + 255) / 256), dim3(256), 0, stream>>>(bufB, an, mbuf, den, nh);
    }
    // 4-6) edge softmax
    {
      const int eh = ETOT * H;
      const dim3 g((eh + 255) / 256), b(256);
      edge_e_kernel   <<<g, b, 0, stream>>>(esrc, edst, NEDGES, ETOT, a_s, a_d,
                                            alphaSec[L], mbuf, H);
      edge_exp_kernel <<<g, b, 0, stream>>>(edst, NEDGES, ETOT, alphaSec[L], mbuf, den, H);
      edge_alpha_kernel<<<g, b, 0, stream>>>(edst, NEDGES, ETOT, alphaSec[L], den, H);
    }
    // 7) weighted scatter-add of source features
    {
      const long long thr = (long long)ETOT * (CH >> 2);
      scatter_kernel<<<dim3((unsigned)((thr + 255) / 256)), dim3(256), 0, stream>>>(
          esrc, edst, NEDGES, ETOT, bufA, alphaSec[L], bufB, H, OC);
    }
    // 8) bias + BN (+ ELU for layers 0,1); layer 2 writes the h output section
    {
      const int nd = NNODES * CH;
      float* hout = (L == 2) ? out : bufB;
      bn_kernel<<<dim3((nd + 255) / 256), dim3(256), 0, stream>>>(
          bufB, Bi[L], Ga[L], Be[L], Rm[L], Rv[L], hout, NNODES, CH, (L < 2) ? 1 : 0);
    }
    hin = bufB;
  }
}